// STEmbedding_81990925680892
// MI455X (gfx1250) — compile-verified
//
#include <hip/hip_runtime.h>
#include <hip/hip_bf16.h>

typedef __attribute__((ext_vector_type(2))) float v2f;
typedef __attribute__((ext_vector_type(8))) float v8f;

#define D64      64
#define NV       325
#define NROWS_TE 1536   // 64 * 24
#define LDS_STRIDE 68   // 64 + 4 pad: avoids bank conflicts on K-major reads

// ---------------------------------------------------------------------------
// Helper: one 16x16 (M,N) tile of C += A(16x64) @ B(64x16) via 16 chained
// V_WMMA_F32_16X16X4_F32. A fragments are preloaded v2f pairs; B is read
// column-strided. All loads issued before the WMMA chain.
// ---------------------------------------------------------------------------
__device__ __forceinline__ v8f gemm16x64_tile(const v2f* __restrict__ afrag,
                                              const float* __restrict__ Bbase)
{
    // Bbase already offset by (2*half)*D64 + n0 + r
    v2f bfrag[16];
    #pragma unroll
    for (int kb = 0; kb < 16; ++kb) {
        bfrag[kb][0] = Bbase[(4 * kb    ) * D64];
        bfrag[kb][1] = Bbase[(4 * kb + 1) * D64];
    }
    v8f acc = {};
    #pragma unroll
    for (int kb = 0; kb < 16; ++kb) {
        acc = __builtin_amdgcn_wmma_f32_16x16x4_f32(false, afrag[kb], false, bfrag[kb],
                                                    (short)0, acc, false, false);
    }
    return acc;
}

// ---------------------------------------------------------------------------
// Kernel 1: se = relu(SE @ W_s1 + b_s1) @ W_s2 + b_s2        (325 x 64)
// One block = 128 threads = 4 waves; block owns a 16-row M stripe.
// Wave w owns N-tile [16w, 16w+16). Hidden 16x64 tile lives in LDS.
// ---------------------------------------------------------------------------
__global__ __launch_bounds__(128)
void se_mlp_kernel(const float* __restrict__ SE,
                   const float* __restrict__ Ws1, const float* __restrict__ bs1,
                   const float* __restrict__ Ws2, const float* __restrict__ bs2,
                   float* __restrict__ se_out)
{
    __shared__ float hid[16 * LDS_STRIDE];

    const int tid  = threadIdx.x;
    const int lane = tid & 31;
    const int wave = tid >> 5;        // 0..3
    const int half = lane >> 4;       // 0 or 1
    const int r    = lane & 15;       // 0..15
    const int m0   = blockIdx.x * 16;
    const int n0   = wave * 16;

    // ---- layer 1: A fragments from global (branchless edge handling) ----
    const int  grow = m0 + r;
    const bool mok  = (grow < NV);
    const int  srow = mok ? grow : 0;          // clamp: always in-bounds
    const float msk = mok ? 1.0f : 0.0f;       // zero the pad rows afterwards

    const float* Ap = SE + srow * D64 + 2 * half;   // 8B-aligned
    v2f afrag[16];
    #pragma unroll
    for (int kb = 0; kb < 16; ++kb) {
        v2f a = *reinterpret_cast<const v2f*>(Ap + 4 * kb);
        a[0] *= msk; a[1] *= msk;
        afrag[kb] = a;
    }

    v8f acc = gemm16x64_tile(afrag, Ws1 + (2 * half) * D64 + n0 + r);

    // bias + relu -> LDS hidden tile
    const float bias1 = bs1[n0 + r];
    #pragma unroll
    for (int j = 0; j < 8; ++j) {
        const int mloc = j + 8 * half;             // 0..15
        hid[mloc * LDS_STRIDE + n0 + r] = fmaxf(acc[j] + bias1, 0.0f);
    }
    __syncthreads();

    // ---- layer 2: A fragments from LDS (8B-aligned ds_load_b64) ----
    const float* Hp = hid + r * LDS_STRIDE + 2 * half;
    v2f afrag2[16];
    #pragma unroll
    for (int kb = 0; kb < 16; ++kb)
        afrag2[kb] = *reinterpret_cast<const v2f*>(Hp + 4 * kb);

    v8f acc2 = gemm16x64_tile(afrag2, Ws2 + (2 * half) * D64 + n0 + r);

    const float bias2 = bs2[n0 + r];
    #pragma unroll
    for (int j = 0; j < 8; ++j) {
        const int mloc = j + 8 * half;
        const int gm   = m0 + mloc;
        if (gm < NV)
            se_out[gm * D64 + n0 + r] = acc2[j] + bias2;
    }
}

// ---------------------------------------------------------------------------
// Kernel 2: te = relu(W_t1[dow] + W_t1[7+tod] + b_t1) @ W_t2 + b_t2
// 1536 rows; one block (128 thr) per 16-row stripe.
// ---------------------------------------------------------------------------
__global__ __launch_bounds__(128)
void te_mlp_kernel(const int* __restrict__ TE, const int* __restrict__ Tptr,
                   const float* __restrict__ Wt1, const float* __restrict__ bt1,
                   const float* __restrict__ Wt2, const float* __restrict__ bt2,
                   float* __restrict__ te_out)
{
    __shared__ float hid[16 * LDS_STRIDE];

    const int tid  = threadIdx.x;
    const int lane = tid & 31;
    const int wave = tid >> 5;
    const int half = lane >> 4;
    const int r    = lane & 15;
    const int row0 = blockIdx.x * 16;
    const int n0   = wave * 16;
    const int T    = Tptr[0];

    // gather + relu -> LDS hidden tile (16 rows x 64)
    // 128 threads: 8 threads per row, 8 contiguous d's each (v2f-able by compiler)
    {
        const int rl  = tid >> 3;          // 0..15 local row
        const int d0  = (tid & 7) * 8;     // 0,8,...,56
        const int gr  = row0 + rl;
        const int dow = TE[2 * gr    ] % 7;
        const int tod = TE[2 * gr + 1] % T;
        const float* w_dow = Wt1 + dow * D64 + d0;
        const float* w_tod = Wt1 + (7 + tod) * D64 + d0;
        const float* bp    = bt1 + d0;
        float* hp = hid + rl * LDS_STRIDE + d0;
        #pragma unroll
        for (int i = 0; i < 8; ++i)
            hp[i] = fmaxf(w_dow[i] + w_tod[i] + bp[i], 0.0f);
    }
    __syncthreads();

    const float* Hp = hid + r * LDS_STRIDE + 2 * half;
    v2f afrag[16];
    #pragma unroll
    for (int kb = 0; kb < 16; ++kb)
        afrag[kb] = *reinterpret_cast<const v2f*>(Hp + 4 * kb);

    v8f acc = gemm16x64_tile(afrag, Wt2 + (2 * half) * D64 + n0 + r);

    const float bias = bt2[n0 + r];
    #pragma unroll
    for (int j = 0; j < 8; ++j) {
        const int mloc = j + 8 * half;
        te_out[(row0 + mloc) * D64 + n0 + r] = acc[j] + bias;
    }
}

// ---------------------------------------------------------------------------
// Kernel 3: out[b,t,v,d] = se[v,d] + te[b*24+t, d]   -- 127.8 MB of writes.
// grid = (6 vertex-chunks of 64, 1536 bt rows), 256 threads, float4 stores.
// This is the roofline-dominant kernel: ~5.5 us at 23.3 TB/s.
// ---------------------------------------------------------------------------
__global__ __launch_bounds__(256)
void broadcast_add_kernel(const float* __restrict__ se,
                          const float* __restrict__ te,
                          float* __restrict__ out)
{
    const int tid   = threadIdx.x;
    const int d4    = tid & 15;          // 16 float4 per 64-wide row
    const int vlane = tid >> 4;          // 0..15
    const int bt    = blockIdx.y;        // 0..1535
    const int v0    = blockIdx.x * 64;

    const float4 tv = reinterpret_cast<const float4*>(te + (size_t)bt * D64)[d4];

    #pragma unroll
    for (int p = 0; p < 4; ++p) {
        const int v = v0 + p * 16 + vlane;
        if (v < NV) {
            const float4 sv = reinterpret_cast<const float4*>(se + (size_t)v * D64)[d4];
            float4 o;
            o.x = sv.x + tv.x; o.y = sv.y + tv.y;
            o.z = sv.z + tv.z; o.w = sv.w + tv.w;
            reinterpret_cast<float4*>(out + ((size_t)bt * NV + v) * D64)[d4] = o;
        }
    }
}

extern "C" void kernel_launch(void* const* d_in, const int* in_sizes, int n_in,
                              void* d_out, int out_size, void* d_ws, size_t ws_size,
                              hipStream_t stream)
{
    const float* SE  = (const float*)d_in[0];
    const int*   TE  = (const int*)  d_in[1];
    const int*   T   = (const int*)  d_in[2];
    const float* Ws1 = (const float*)d_in[3];
    const float* bs1 = (const float*)d_in[4];
    const float* Ws2 = (const float*)d_in[5];
    const float* bs2 = (const float*)d_in[6];
    const float* Wt1 = (const float*)d_in[7];
    const float* bt1 = (const float*)d_in[8];
    const float* Wt2 = (const float*)d_in[9];
    const float* bt2 = (const float*)d_in[10];
    float* out = (float*)d_out;

    float* se_ws = (float*)d_ws;                 // 325*64 floats (83,200 B)
    float* te_ws = se_ws + NV * D64;             // 1536*64 floats (393,216 B)

    se_mlp_kernel<<<dim3(21), dim3(128), 0, stream>>>(SE, Ws1, bs1, Ws2, bs2, se_ws);
    te_mlp_kernel<<<dim3(NROWS_TE / 16), dim3(128), 0, stream>>>(TE, T, Wt1, bt1, Wt2, bt2, te_ws);
    broadcast_add_kernel<<<dim3(6, NROWS_TE), dim3(256), 0, stream>>>(se_ws, te_ws, out);

    (void)in_sizes; (void)n_in; (void)out_size; (void)ws_size;
}